// Project2D3Dv2_23252952941255
// MI455X (gfx1250) — compile-verified
//
#include <hip/hip_runtime.h>

// ---------------------------------------------------------------------------
// Project2D3D for MI455X (gfx1250, wave32).
// Pure bandwidth problem: 265 MB stores + ~40 MB gathered reads -> ~13 us at
// 23.3 TB/s. No matrix math; optimize the data-movement path instead:
//  * phase 1: tiny VALU kernel -> packed per-voxel pixel index (or -1)
//  * phase 2: per-block index staging through LDS via CDNA5 async
//    global->LDS ops (ASYNCcnt), then coalesced gather/scatter with
//    global_prefetch_b8 gather-ahead.
//  * fallback fused kernel (no workspace) if ws_size is too small.
// ---------------------------------------------------------------------------

#define SX 60
#define SY 36
#define SZ 60
#define N_VOX (SX * SY * SZ)      // 129600
#define IMG_W 640
#define IMG_H 480
#define DSW 160                   // IMG_W / 4
#define DSH 120                   // IMG_H / 4
#define HW (DSW * DSH)            // 19200 elements per channel plane
#define C_TOT 512
#define C_CHUNK 32
#define BLK 256

#if defined(__has_builtin)
#  if __has_builtin(__builtin_amdgcn_global_load_async_to_lds_b32)
#    define HAVE_ASYNC_LDS 1
#  endif
#endif

typedef __attribute__((address_space(1))) int* gptr_i32;
typedef __attribute__((address_space(3))) int* lptr_i32;

// ---------------------------------------------------------------------------
// Shared projection math. Matches reference exactly: rintf == jnp.round
// (round-half-even under default RNE), float frustum compares, fliplr before
// the /4 downsample. Returns pixel index >= 0 or -1 when culled.
// ---------------------------------------------------------------------------
__device__ __forceinline__ int project_voxel(int i,
                                             const float* __restrict__ invP,
                                             const float* __restrict__ vorig,
                                             const float* __restrict__ camK,
                                             int flip)
{
    const int gx = i / (SY * SZ);
    const int r  = i - gx * (SY * SZ);
    const int gy = r / SZ;
    const int gz = r - gy * SZ;

    const float wx = gx * 0.08f + 0.04f;
    const float wy = gy * 0.08f + 0.04f;
    const float wz = gz * 0.08f + 0.04f;

    // pts_world = (wx, wz, wy) + vox_origin   (note the y/z swap in reference)
    const float px = wx + vorig[0];
    const float py = wz + vorig[1];
    const float pz = wy + vorig[2];

    // pts_cam = inv_cam_pose @ [p,1]
    const float cx = invP[0] * px + invP[1] * py + invP[2]  * pz + invP[3];
    const float cy = invP[4] * px + invP[5] * py + invP[6]  * pz + invP[7];
    const float cz = invP[8] * px + invP[9] * py + invP[10] * pz + invP[11];

    // pts_img = cam_k @ pts_cam
    const float ix = camK[0] * cx + camK[1] * cy + camK[2] * cz;
    const float iy = camK[3] * cx + camK[4] * cy + camK[5] * cz;
    const float iz = camK[6] * cx + camK[7] * cy + camK[8] * cz;

    const float u = rintf(ix / iz);   // jnp.round
    const float v = rintf(iy / iz);

    const bool mask = (cz > 0.0f) &&
                      (u >= 0.0f) && (u < (float)IMG_W) &&
                      (v >= 0.0f) && (v < (float)IMG_H);

    int col = (int)u;                 // in [0, 639] whenever mask is true
    int row = (int)v;                 // in [0, 479] whenever mask is true
    if (flip) col = (IMG_W - 1) - col;
    row >>= 2;                        // floor(row / DS), row >= 0 under mask
    col >>= 2;
    const int lin = row * DSW + col;

    return mask ? lin : -1;
}

// ---------------------------------------------------------------------------
// Shared gather/scatter inner loop for one voxel column x C_CHUNK channels.
// Stores are coalesced across lanes (consecutive i).
// ---------------------------------------------------------------------------
__device__ __forceinline__ void gather_column(const float* __restrict__ x2d,
                                              float* __restrict__ out,
                                              int i, int c0, int p)
{
    float* __restrict__ dst = out + (size_t)c0 * N_VOX + i;

    if (p < 0) {
        // Culled voxel: dense zero column (output buffer is poisoned, every
        // element must be written).
#pragma unroll
        for (int cc = 0; cc < C_CHUNK; ++cc) {
            *dst = 0.0f;
            dst += N_VOX;
        }
    } else {
        const float* __restrict__ src = x2d + (size_t)c0 * HW + p;
        // Gather-ahead prefetch: each lane prefetches its own column 8 channel
        // planes (8 * 76.8 KB) ahead. Skip on the last chunk to keep every
        // prefetch address inside the x2d allocation.
        const bool doPf = (c0 + C_CHUNK + 8) <= C_TOT;
#pragma unroll 8
        for (int cc = 0; cc < C_CHUNK; ++cc) {
            if (doPf) __builtin_prefetch(src + 8 * HW, 0, 0);  // global_prefetch_b8
            const float fv = *src;
            *dst = fv;
            src += HW;
            dst += N_VOX;
        }
    }
}

// ---------------------------------------------------------------------------
// Phase 1: per-voxel projection -> packed pixel index (or -1) in workspace.
// ---------------------------------------------------------------------------
__global__ void proj_idx_kernel(const float* __restrict__ invP,
                                const float* __restrict__ vorig,
                                const float* __restrict__ camK,
                                const int*   __restrict__ fliplr,
                                int* __restrict__ idx_out)
{
    const int i = blockIdx.x * blockDim.x + threadIdx.x;
    if (i >= N_VOX) return;
    idx_out[i] = project_voxel(i, invP, vorig, camK, fliplr[0]);
}

// ---------------------------------------------------------------------------
// Phase 2: gather x2d[c, p_i] -> out[c, i] for a 256-voxel x 32-channel tile.
// The 256 indices are staged once into LDS through the CDNA5 async data mover
// and reused for all 32 channels of the chunk.
// ---------------------------------------------------------------------------
__global__ void gather_scatter_kernel(const float* __restrict__ x2d,
                                      const int* __restrict__ vidx,
                                      float* __restrict__ out)
{
    __shared__ int sIdx[BLK];

    const int  i     = blockIdx.x * BLK + threadIdx.x;
    const bool valid = (i < N_VOX);

#if defined(HAVE_ASYNC_LDS)
    // CDNA5 async global->LDS copy, tracked with ASYNCcnt.
    if (valid) {
        __builtin_amdgcn_global_load_async_to_lds_b32(
            (gptr_i32)(vidx + i),
            (lptr_i32)(&sIdx[threadIdx.x]),
            /*offset=*/0, /*cpol=*/0);
    }
#  if __has_builtin(__builtin_amdgcn_s_wait_asynccnt)
    __builtin_amdgcn_s_wait_asynccnt(0);
#  else
    asm volatile("s_wait_asynccnt 0x0" ::: "memory");
#  endif
#else
    if (valid) sIdx[threadIdx.x] = vidx[i];
#endif
    __syncthreads();
    if (!valid) return;

    gather_column(x2d, out, i, blockIdx.y * C_CHUNK, sIdx[threadIdx.x]);
}

// ---------------------------------------------------------------------------
// Fallback: fused projection + gather (used only when d_ws is too small to
// hold the index array). Recomputes the cheap projection per channel chunk.
// ---------------------------------------------------------------------------
__global__ void fused_kernel(const float* __restrict__ x2d,
                             const float* __restrict__ invP,
                             const float* __restrict__ vorig,
                             const float* __restrict__ camK,
                             const int*   __restrict__ fliplr,
                             float* __restrict__ out)
{
    const int i = blockIdx.x * BLK + threadIdx.x;
    if (i >= N_VOX) return;
    const int p = project_voxel(i, invP, vorig, camK, fliplr[0]);
    gather_column(x2d, out, i, blockIdx.y * C_CHUNK, p);
}

// ---------------------------------------------------------------------------
// Launch. Inputs (setup_inputs order):
//   0: x2d (512*120*160 f32)  1: inv_cam_pose (16 f32)  2: vox_origin (3 f32)
//   3: cam_k (9 f32)          4: old_voxel_indices (unused)
//   5: old_image_indices (unused)                      6: fliplr (1 i32)
// d_out: 512*129600 f32. d_ws: N_VOX ints of scratch for packed indices.
// ---------------------------------------------------------------------------
extern "C" void kernel_launch(void* const* d_in, const int* in_sizes, int n_in,
                              void* d_out, int out_size, void* d_ws, size_t ws_size,
                              hipStream_t stream)
{
    (void)in_sizes; (void)n_in; (void)out_size;

    const float* x2d   = (const float*)d_in[0];
    const float* invP  = (const float*)d_in[1];
    const float* vorig = (const float*)d_in[2];
    const float* camK  = (const float*)d_in[3];
    const int*   flip  = (const int*)d_in[6];

    float* out = (float*)d_out;

    const int nb = (N_VOX + BLK - 1) / BLK;   // 507 blocks over voxels
    dim3 grid(nb, C_TOT / C_CHUNK);           // 507 x 16

    if (ws_size >= (size_t)N_VOX * sizeof(int)) {
        int* vidx = (int*)d_ws;
        proj_idx_kernel<<<nb, BLK, 0, stream>>>(invP, vorig, camK, flip, vidx);
        gather_scatter_kernel<<<grid, BLK, 0, stream>>>(x2d, vidx, out);
    } else {
        fused_kernel<<<grid, BLK, 0, stream>>>(x2d, invP, vorig, camK, flip, out);
    }
}